// meanPred_57011395887639
// MI455X (gfx1250) — compile-verified
//
#include <hip/hip_runtime.h>
#include <hip/hip_bf16.h>

// Problem constants from the reference (fixed sizes).
#define T_ 16
#define P_ 4096
#define Q_ 2048
#define D_ 512

typedef __attribute__((ext_vector_type(2))) float v2f;
typedef __attribute__((ext_vector_type(8))) float v8f;

// ---------------------------------------------------------------------------
// Kernel 1: per-t counts of pos/neg kept pool entries.
// pos_keep = (mask != 0) && (label != 0) ; neg_keep = (mask != 0) && (label != 1)
// ---------------------------------------------------------------------------
__global__ void count_kernel(const int* __restrict__ labels,
                             const int* __restrict__ mask,
                             int* __restrict__ cnt /* [2*T_]: pos then neg */) {
    const int t   = blockIdx.x;
    const int tid = threadIdx.x;
    int cp = 0, cn = 0;
    for (int p = tid; p < P_; p += 256) {
        const int l = labels[t * P_ + p];
        const int m = mask[t * P_ + p];
        if (m != 0) {
            if (l != 0) cp++;
            if (l != 1) cn++;
        }
    }
    __shared__ int sp[256];
    __shared__ int sn[256];
    sp[tid] = cp; sn[tid] = cn;
    __syncthreads();
    for (int s = 128; s > 0; s >>= 1) {
        if (tid < s) { sp[tid] += sp[tid + s]; sn[tid] += sn[tid + s]; }
        __syncthreads();
    }
    if (tid == 0) { cnt[t] = sp[0]; cnt[T_ + t] = sn[0]; }
}

// ---------------------------------------------------------------------------
// Kernel 2: fused row-norm + signed-weight + weighted column accumulation.
// One block handles 16 contiguous pool rows of one t (32KB LDS tile, read once
// from HBM -> this is the 134MB stream). w[t,d] += sum_r coeff_r/||row_r|| * row_r[d].
// ---------------------------------------------------------------------------
__global__ void pool_accum_kernel(const float* __restrict__ pool,
                                  const int* __restrict__ labels,
                                  const int* __restrict__ mask,
                                  const int* __restrict__ cnt,
                                  float* __restrict__ w /* [T_*D_] */) {
    __shared__ float rows[16][D_];   // 32 KB
    __shared__ float rw[16];

    const int b     = blockIdx.x;          // 0 .. T_*(P_/16)-1
    const int t     = b >> 8;              // P_/16 = 256 chunks per t
    const int pbase = (b & 255) << 4;
    const int tid   = threadIdx.x;
    const int lane  = tid & 31;
    const int wv    = tid >> 5;            // 8 waves

    // Coalesced contiguous load: 16 rows * 512 f32 = 8192 floats = 2048 float4
    const float4* src = (const float4*)(pool + ((size_t)t * P_ + pbase) * D_);
    float4* dst = (float4*)&rows[0][0];
    for (int i = tid; i < (16 * D_) / 4; i += 256) dst[i] = src[i];
    __syncthreads();

    const float invPos = 1.0f / (float)max(cnt[t], 1);
    const float invNeg = 1.0f / (float)max(cnt[T_ + t], 1);

    // Each wave computes norms/weights for 2 rows.
    for (int rr = 0; rr < 2; ++rr) {
        const int r = wv * 2 + rr;
        float s = 0.0f;
        for (int c = lane; c < D_; c += 32) {
            const float v = rows[r][c];
            s += v * v;
        }
        for (int off = 16; off > 0; off >>= 1) s += __shfl_xor(s, off);
        if (lane == 0) {
            const int l = labels[t * P_ + pbase + r];
            const int m = mask[t * P_ + pbase + r];
            float coeff = 0.0f;
            if (m != 0) {
                if (l != 0) coeff = invPos;        // pos_keep
                else if (l != 1) coeff = -invNeg;  // neg_keep (l==0 here)
            }
            rw[r] = (s > 0.0f) ? coeff * rsqrtf(s) : 0.0f;
        }
    }
    __syncthreads();

    // Column reduction over the 16 staged rows; 2 columns per thread.
    float a0 = 0.0f, a1 = 0.0f;
    #pragma unroll
    for (int r = 0; r < 16; ++r) {
        const float c = rw[r];
        a0 += c * rows[r][tid];
        a1 += c * rows[r][tid + 256];
    }
    atomicAdd(&w[t * D_ + tid], a0);
    atomicAdd(&w[t * D_ + tid + 256], a1);
}

// ---------------------------------------------------------------------------
// Kernel 3: pred[t,q] = dot(query[t,q], w[t]) * rsqrt(||query[t,q]||^2)
// via V_WMMA_F32_16X16X4_F32. Each wave owns a 16-query tile; w is broadcast
// across all 16 B columns so every output column carries the same dot.
// A layout (ISA 16x4 f32): lanes 0-15 hold K0(v0)/K1(v1), lanes 16-31 K2/K3.
// ---------------------------------------------------------------------------
__global__ void gemv_wmma_kernel(const float* __restrict__ query,
                                 const float* __restrict__ w,
                                 float* __restrict__ pred /* [T_*Q_] */) {
    __shared__ float ldsW[D_];        // 2 KB  (per-t weight vector)
    __shared__ float ldsQ[8][16 * 64];// 32 KB (per-wave query tile chunk)
    __shared__ float ldsDot[8][16];

    const int b     = blockIdx.x;     // 0 .. T_*Q_/128 - 1  (= 256)
    const int t     = b >> 4;         // Q_/128 = 16 blocks per t
    const int qblk  = (b & 15) << 7;  // 128 queries per block
    const int tid   = threadIdx.x;
    const int lane  = tid & 31;
    const int wv    = tid >> 5;
    const int qbase = qblk + wv * 16;
    const int m     = lane & 15;
    const int off   = (lane < 16) ? 0 : 2;

    for (int i = tid; i < D_; i += 256) ldsW[i] = w[t * D_ + i];
    __syncthreads();

    v8f acc = {};
    float ssq = 0.0f;

    for (int c = 0; c < D_; c += 64) {
        // Stage 16 rows x 64 cols of this wave's queries (float2, coalesced).
        #pragma unroll 4
        for (int r = 0; r < 16; ++r) {
            const float2 v = *(const float2*)(query +
                ((size_t)t * Q_ + qbase + r) * D_ + c + 2 * lane);
            *(float2*)&ldsQ[wv][r * 64 + 2 * lane] = v;
        }
        __syncthreads();   // uniform across block; makes LDS writes lane-visible

        #pragma unroll
        for (int k = 0; k < 64; k += 4) {
            v2f a, bb;
            a[0]  = ldsQ[wv][m * 64 + k + off];
            a[1]  = ldsQ[wv][m * 64 + k + off + 1];
            bb[0] = ldsW[c + k + off];
            bb[1] = ldsW[c + k + off + 1];
            ssq += a[0] * a[0] + a[1] * a[1];
            acc = __builtin_amdgcn_wmma_f32_16x16x4_f32(
                /*neg_a=*/false, a, /*neg_b=*/false, bb,
                /*c_mod=*/(short)0, acc, /*reuse_a=*/false, /*reuse_b=*/false);
        }
        __syncthreads();
    }

    // Full row sum-of-squares: combine lane L with lane L^16.
    ssq += __shfl_xor(ssq, 16);

    // All output columns are identical (broadcast B). C/D layout:
    // lane 0: rows 0-7 in acc[0..7]; lane 16: rows 8-15 in acc[0..7].
    if (lane == 0) {
        #pragma unroll
        for (int i = 0; i < 8; ++i) ldsDot[wv][i] = acc[i];
    }
    if (lane == 16) {
        #pragma unroll
        for (int i = 0; i < 8; ++i) ldsDot[wv][8 + i] = acc[i];
    }
    __syncthreads();

    if (lane < 16) {
        pred[(size_t)t * Q_ + qbase + lane] = ldsDot[wv][lane] * rsqrtf(ssq);
    }
}

// ---------------------------------------------------------------------------
extern "C" void kernel_launch(void* const* d_in, const int* in_sizes, int n_in,
                              void* d_out, int out_size, void* d_ws, size_t ws_size,
                              hipStream_t stream) {
    const float* pool   = (const float*)d_in[0];
    const float* query  = (const float*)d_in[1];
    const int*   labels = (const int*)d_in[2];
    const int*   mask   = (const int*)d_in[3];
    float*       pred   = (float*)d_out;

    float* w   = (float*)d_ws;                                  // T_*D_ floats
    int*   cnt = (int*)((char*)d_ws + T_ * D_ * sizeof(float)); // 2*T_ ints

    // w is accumulated with atomics; must be zeroed every call (graph-safe).
    hipMemsetAsync(w, 0, T_ * D_ * sizeof(float), stream);

    count_kernel<<<T_, 256, 0, stream>>>(labels, mask, cnt);
    pool_accum_kernel<<<T_ * (P_ / 16), 256, 0, stream>>>(pool, labels, mask, cnt, w);
    gemv_wmma_kernel<<<(T_ * Q_) / 128, 256, 0, stream>>>(query, w, pred);
}